// QuantizationLayer_60232621359584
// MI455X (gfx1250) — compile-verified
//
#include <hip/hip_runtime.h>
#include <hip/hip_bf16.h>

// QuantizationLayer bit-unpack: out[b, n*3+i] = bit (2-i) of round(x*8 - 0.5).
// Pure HBM-streaming kernel (512 MiB total traffic -> ~23us floor at 23.3 TB/s).
// Strategy: b128 NT loads, b128 NT stores, v_rndne_f32 for exact round-half-even,
// global_prefetch_b8 ahead of the read stream. One 4-float vector of input -> 3
// output vectors (12 contiguous floats) per thread, so a wave's stores tile a
// contiguous 1536B span and write-combine cleanly in L2.

// Clang native vector type: required by __builtin_nontemporal_* (HIP's float4
// is a struct and is rejected by the builtin).
typedef __attribute__((ext_vector_type(4))) float v4f;

#ifndef PF_DIST_VEC4
#define PF_DIST_VEC4 (16384)  // prefetch 16384 vec4 = 256 KiB ahead of the load stream
#endif

__device__ __forceinline__ float bit_f(int q, int sh) {
  return (float)((q >> sh) & 1);
}

__global__ __launch_bounds__(256) void quant_unpack3_kernel(
    const v4f* __restrict__ in, v4f* __restrict__ out, int n4) {
  int i = blockIdx.x * blockDim.x + threadIdx.x;
  if (i >= n4) return;

  // gfx1250 global_prefetch_b8 ahead of the read stream (speculative; cheap).
  if (i + PF_DIST_VEC4 < n4) {
    __builtin_prefetch((const void*)(in + i + PF_DIST_VEC4), 0, 0);
  }

  v4f x = __builtin_nontemporal_load(&in[i]);

  // q = rint(x*8 - 0.5): v_rndne_f32 == round-half-to-even == jnp.round.
  // x*8 is exact (pow2 scale), so fma == (x*8)-0.5 bit-exactly. q in [0,7].
  int q0 = (int)rintf(__builtin_fmaf(x.x, 8.0f, -0.5f));
  int q1 = (int)rintf(__builtin_fmaf(x.y, 8.0f, -0.5f));
  int q2 = (int)rintf(__builtin_fmaf(x.z, 8.0f, -0.5f));
  int q3 = (int)rintf(__builtin_fmaf(x.w, 8.0f, -0.5f));

  // 12 output floats, MSB-first per element, packed into 3 vec4s.
  v4f o0, o1, o2;
  o0.x = bit_f(q0, 2); o0.y = bit_f(q0, 1); o0.z = bit_f(q0, 0); o0.w = bit_f(q1, 2);
  o1.x = bit_f(q1, 1); o1.y = bit_f(q1, 0); o1.z = bit_f(q2, 2); o1.w = bit_f(q2, 1);
  o2.x = bit_f(q2, 0); o2.y = bit_f(q3, 2); o2.z = bit_f(q3, 1); o2.w = bit_f(q3, 0);

  v4f* ob = out + (size_t)i * 3;
  __builtin_nontemporal_store(o0, ob + 0);
  __builtin_nontemporal_store(o1, ob + 1);
  __builtin_nontemporal_store(o2, ob + 2);
}

extern "C" void kernel_launch(void* const* d_in, const int* in_sizes, int n_in,
                              void* d_out, int out_size, void* d_ws, size_t ws_size,
                              hipStream_t stream) {
  (void)n_in; (void)out_size; (void)d_ws; (void)ws_size;
  const v4f* x = (const v4f*)d_in[0];
  v4f* out = (v4f*)d_out;

  int n = in_sizes[0];          // 65536*512 = 33,554,432 floats (divisible by 4)
  int n4 = n >> 2;              // 8,388,608 vec4s
  const int block = 256;        // 8 wave32s per block
  int grid = (n4 + block - 1) / block;

  quant_unpack3_kernel<<<grid, block, 0, stream>>>(x, out, n4);
}